// Attention_86199993631321
// MI455X (gfx1250) — compile-verified
//
#include <hip/hip_runtime.h>

typedef float v2f __attribute__((ext_vector_type(2)));
typedef float v4f __attribute__((ext_vector_type(4)));
typedef float v8f __attribute__((ext_vector_type(8)));

#define B_DIM 8
#define N_DIM 128
#define H_DIM 512
#define M_DIM (B_DIM * N_DIM)   // 1024 rows of Y

static __device__ __forceinline__ v8f wmma4(v2f a, v2f b, v8f c) {
  return __builtin_amdgcn_wmma_f32_16x16x4_f32(
      /*neg_a=*/false, a, /*neg_b=*/false, b,
      /*c_mod=*/(short)0, c, /*reuse_a=*/false, /*reuse_b=*/false);
}

// ---------------------------------------------------------------------------
// Kernel 1: Y[m][o] = sum_h X[m][h] * W[o][h]   (X: 1024x512, W: 512x512)
// One wave32 computes a 32x32 tile of Y (2x2 blocking of 16x16 WMMA tiles):
// per k-step, 2 A-loads + 2 B-loads feed 4 v_wmma_f32_16x16x4_f32 ops,
// halving L2 traffic vs. unblocked (64 MB instead of 128 MB).
// A (16x4 f32) layout:  v0 = K=0 (lanes 0-15), K=2 (lanes 16-31)
//                       v1 = K=1 (lanes 0-15), K=3 (lanes 16-31)
// B (4x16 f32) layout:  v0 = row K=0 (lanes = N), row K=2 (lanes 16-31)
//                       v1 = row K=1,             row K=3
// Both are contiguous float2 loads along h for this problem.
// ---------------------------------------------------------------------------
__global__ __launch_bounds__(256) void gemm_xwT_kernel(
    const float* __restrict__ X, const float* __restrict__ W,
    float* __restrict__ Y) {
  const int wave = (blockIdx.x * blockDim.x + threadIdx.x) >> 5;  // 0..511
  const int lane = threadIdx.x & 31;
  const int tile_n = wave & 15;   // 512/32 = 16 tiles across O
  const int tile_m = wave >> 4;   // 1024/32 = 32 tiles down M
  const int m0 = tile_m << 5;
  const int n0 = tile_n << 5;
  const int half = lane >> 4;     // 0: K pair {0,1}, 1: K pair {2,3}
  const int l16  = lane & 15;

  const float* A0 = X + (size_t)(m0 + l16) * H_DIM + 2 * half;
  const float* A1 = A0 + (size_t)16 * H_DIM;
  const float* B0 = W + (size_t)(n0 + l16) * H_DIM + 2 * half;
  const float* B1 = B0 + (size_t)16 * H_DIM;

  v8f c00 = {}, c01 = {}, c10 = {}, c11 = {};
#pragma unroll 4
  for (int k = 0; k < H_DIM; k += 4) {
    v2f a0 = *(const v2f*)(A0 + k);
    v2f a1 = *(const v2f*)(A1 + k);
    v2f b0 = *(const v2f*)(B0 + k);
    v2f b1 = *(const v2f*)(B1 + k);
    c00 = wmma4(a0, b0, c00);
    c01 = wmma4(a0, b1, c01);
    c10 = wmma4(a1, b0, c10);
    c11 = wmma4(a1, b1, c11);
  }

  // D layout: c[r] -> row (tile_m0 + r + 8*half), col (tile_n0 + l16)
  float* o0 = Y + (size_t)(m0 + 8 * half) * H_DIM + n0 + l16;
  float* o1 = o0 + (size_t)16 * H_DIM;
#pragma unroll
  for (int r = 0; r < 8; ++r) {
    o0[(size_t)r * H_DIM]      = c00[r];
    o0[(size_t)r * H_DIM + 16] = c01[r];
    o1[(size_t)r * H_DIM]      = c10[r];
    o1[(size_t)r * H_DIM + 16] = c11[r];
  }
}

// ---------------------------------------------------------------------------
// Kernel 2 (bandwidth-bound part): out[b,i,j,o] = Y[b,i,o] + Y[b,j,o] + bias[o]
// blockIdx.x = b*N + i ; blockIdx.y selects a chunk of 16 j values.
// 128 threads cover the 512-float o dimension as v4f (b128).
// yi+bias kept in registers; Y rows (2 MB total) stay hot in L2;
// the 256 MB output uses non-temporal b128 stores to not thrash L2.
// ---------------------------------------------------------------------------
__global__ __launch_bounds__(128) void pair_linear_bcast_kernel(
    const float* __restrict__ Y, const float* __restrict__ bias,
    float* __restrict__ out) {
  const int bi = blockIdx.x;          // m = b*N + i  (0..1023)
  const int b  = bi >> 7;             // batch
  const int jc = blockIdx.y << 4;     // j chunk base (16 j's per block)
  const int t  = threadIdx.x;         // v4f index over H (0..127)

  const v4f* __restrict__ Y4 = (const v4f*)Y;
  v4f yi = Y4[(size_t)bi * (H_DIM / 4) + t];
  v4f bv = ((const v4f*)bias)[t];
  yi += bv;

  v4f* O4 = (v4f*)out;
  const size_t jrow = (size_t)b * N_DIM + jc;                         // Y row of (b, jc)
  const size_t obase = ((size_t)bi * N_DIM + jc) * (H_DIM / 4) + t;   // out v4f index

#pragma unroll 4
  for (int jj = 0; jj < 16; ++jj) {
    v4f yj = Y4[(jrow + jj) * (H_DIM / 4) + t];
    v4f r = yi + yj;
    __builtin_nontemporal_store(r, &O4[obase + (size_t)jj * (H_DIM / 4)]);
  }
}

extern "C" void kernel_launch(void* const* d_in, const int* in_sizes, int n_in,
                              void* d_out, int out_size, void* d_ws, size_t ws_size,
                              hipStream_t stream) {
  const float* local = (const float*)d_in[0];  // [8,128,512] f32
  const float* W     = (const float*)d_in[1];  // [512,512]   f32
  const float* bias  = (const float*)d_in[2];  // [512]       f32
  float* Y   = (float*)d_ws;                   // 1024*512*4 = 2 MB scratch
  float* out = (float*)d_out;                  // 256 MB

  // GEMM: 512 waves = 32x16 tiles of 32x32; 8 waves/block -> 64 blocks.
  gemm_xwT_kernel<<<dim3(64), dim3(256), 0, stream>>>(local, W, Y);

  // Broadcast add: 1024 (b,i) rows x 8 j-chunks; 128 threads each.
  pair_linear_bcast_kernel<<<dim3(M_DIM, N_DIM / 16), dim3(128), 0, stream>>>(
      Y, bias, out);
}